// SAGEConv_52458730553708
// MI455X (gfx1250) — compile-verified
//
#include <hip/hip_runtime.h>
#include <hip/hip_bf16.h>

typedef __attribute__((ext_vector_type(2))) float v2f;
typedef __attribute__((ext_vector_type(8))) float v8f;

#define IN_FEATS 128
#define OUT_FEATS 128

// ---------------------------------------------------------------------------
// Kernel 1: zero the workspace (agg [n*128] followed by deg [n]) with float4.
// ---------------------------------------------------------------------------
__global__ void sage_zero_kernel(float4* __restrict__ p, int n4) {
    int i = blockIdx.x * blockDim.x + threadIdx.x;
    if (i < n4) p[i] = make_float4(0.f, 0.f, 0.f, 0.f);
}

// ---------------------------------------------------------------------------
// Kernel 2: edge scatter. One wave (32 lanes) per edge; lane handles 4
// consecutive features (float4 gather + 4 hardware f32 atomics). src/dst are
// wave-uniform loads. Lane 0 also bumps the in-degree counter.
// ---------------------------------------------------------------------------
__global__ void sage_scatter_kernel(const float* __restrict__ x,
                                    const int* __restrict__ esrc,
                                    const int* __restrict__ edst,
                                    float* __restrict__ agg,
                                    float* __restrict__ deg,
                                    int n_edges) {
    int gid = blockIdx.x * blockDim.x + threadIdx.x;
    int e = gid >> 5;            // all 32 lanes of a wave share one edge
    if (e >= n_edges) return;
    int part = gid & 31;         // lane: 4-feature chunk
    int s = esrc[e];
    int d = edst[e];
    const float4 v = *(const float4*)(x + (size_t)s * IN_FEATS + part * 4);
    float* ap = agg + (size_t)d * IN_FEATS + part * 4;
    unsafeAtomicAdd(ap + 0, v.x);   // -> global_atomic_add_f32
    unsafeAtomicAdd(ap + 1, v.y);
    unsafeAtomicAdd(ap + 2, v.z);
    unsafeAtomicAdd(ap + 3, v.w);
    if (part == 0) unsafeAtomicAdd(deg + d, 1.0f);
}

// ---------------------------------------------------------------------------
// Kernel 3: deg -> 1/max(deg,1)
// ---------------------------------------------------------------------------
__global__ void sage_rdeg_kernel(float* __restrict__ deg, int n) {
    int i = blockIdx.x * blockDim.x + threadIdx.x;
    if (i < n) deg[i] = 1.0f / fmaxf(deg[i], 1.0f);
}

// ---------------------------------------------------------------------------
// Kernel 4: fused dual GEMM with fp32 WMMA (V_WMMA_F32_16X16X4_F32).
//   out = (agg * rdeg) @ W + x @ Wr + bias
// One wave computes one 16x16 output tile. 8 waves/block cover all 128 output
// cols; blockIdx.x is the M tile (40000/16 = 2500 blocks). EXEC is all ones.
//
// ISA VGPR layouts (CDNA5 ISA):
//   A (16x4 f32):  lane l, vgpr v -> A[l%16][2*(l/16)+v]   (float2 load)
//   B (4x16 f32):  lane l, vgpr v -> B[2*(l/16)+v][l%16]
//   C/D (16x16):   lane l, vgpr r -> C[r + 8*(l/16)][l%16]
// ---------------------------------------------------------------------------
__global__ __launch_bounds__(256) void sage_gemm_kernel(
        const float* __restrict__ agg, const float* __restrict__ rdeg,
        const float* __restrict__ x,   const float* __restrict__ W,
        const float* __restrict__ Wr,  const float* __restrict__ bias,
        float* __restrict__ out) {
    const int wave = threadIdx.x >> 5;    // 0..7 -> N tile
    const int lane = threadIdx.x & 31;
    const int m0 = blockIdx.x << 4;
    const int n0 = wave << 4;
    const int r  = lane & 15;             // A row / B,C column
    const int kh = lane >> 4;             // K half select (0/1)

    const float scale = rdeg[m0 + r];     // fold mean-normalization into A
    const float bl    = bias[n0 + r];     // fold bias into the store

    const float* aggp = agg + (size_t)(m0 + r) * IN_FEATS + 2 * kh;
    const float* xp   = x   + (size_t)(m0 + r) * IN_FEATS + 2 * kh;
    const float* wp   = W   + (size_t)(2 * kh) * OUT_FEATS + n0 + r;
    const float* wrp  = Wr  + (size_t)(2 * kh) * OUT_FEATS + n0 + r;

    v8f acc = {};
    #pragma unroll 4
    for (int k = 0; k < IN_FEATS; k += 4) {
        v2f a0 = *(const v2f*)(aggp + k);
        a0 = a0 * scale;
        v2f a1 = *(const v2f*)(xp + k);
        v2f b0, b1;
        b0.x = wp[(size_t)k * OUT_FEATS];
        b0.y = wp[(size_t)(k + 1) * OUT_FEATS];
        b1.x = wrp[(size_t)k * OUT_FEATS];
        b1.y = wrp[(size_t)(k + 1) * OUT_FEATS];
        acc = __builtin_amdgcn_wmma_f32_16x16x4_f32(
                  false, a0, false, b0, (short)0, acc, false, false);
        acc = __builtin_amdgcn_wmma_f32_16x16x4_f32(
                  false, a1, false, b1, (short)0, acc, false, false);
    }

    float* op = out + (size_t)(m0 + 8 * kh) * OUT_FEATS + n0 + r;
    #pragma unroll
    for (int i = 0; i < 8; ++i)
        op[(size_t)i * OUT_FEATS] = acc[i] + bl;
}

// ---------------------------------------------------------------------------
extern "C" void kernel_launch(void* const* d_in, const int* in_sizes, int n_in,
                              void* d_out, int out_size, void* d_ws, size_t ws_size,
                              hipStream_t stream) {
    const float* x    = (const float*)d_in[0];
    const int*   src  = (const int*)d_in[1];
    const int*   dst  = (const int*)d_in[2];
    const float* W    = (const float*)d_in[3];
    const float* Wr   = (const float*)d_in[4];
    const float* bias = (const float*)d_in[5];
    float* out = (float*)d_out;

    const int n_nodes = in_sizes[0] / IN_FEATS;   // 40000
    const int n_edges = in_sizes[1];              // 640000

    float* agg = (float*)d_ws;                    // [n_nodes * 128]
    float* deg = agg + (size_t)n_nodes * IN_FEATS; // [n_nodes]

    // 1) zero agg + deg (contiguous)
    int n4 = (n_nodes * IN_FEATS + n_nodes) / 4;
    sage_zero_kernel<<<(n4 + 255) / 256, 256, 0, stream>>>((float4*)agg, n4);

    // 2) scatter-add features + degrees (one wave per edge)
    long long sthreads = (long long)n_edges * 32;
    int sblocks = (int)((sthreads + 255) / 256);
    sage_scatter_kernel<<<sblocks, 256, 0, stream>>>(x, src, dst, agg, deg, n_edges);

    // 3) reciprocal clamped degree
    sage_rdeg_kernel<<<(n_nodes + 255) / 256, 256, 0, stream>>>(deg, n_nodes);

    // 4) fused dual fp32-WMMA GEMM + bias
    sage_gemm_kernel<<<n_nodes / 16, 256, 0, stream>>>(agg, deg, x, W, Wr, bias, out);
}